// EmbertLoss_22728966930830
// MI455X (gfx1250) — compile-verified
//
#include <hip/hip_runtime.h>
#include <math.h>
#include <stdint.h>

// ---------------------------------------------------------------------------
// EmbertLoss on gfx1250 (MI455X): bandwidth-bound streaming top-k per row.
// CDNA5 async global->LDS copies (global_load_async_to_lds_b128, non-temporal)
// with split s_wait_asynccnt waits in a double-buffered pipeline.
// Roofline: 2048 x 50257 x 4B = 412 MB single pass @ 23.3 TB/s => ~17.7 us.
// ---------------------------------------------------------------------------

constexpr int TPB         = 256;   // 8 wave32 per block
constexpr int TILE_FLOATS = 4096;  // 16 KB tile
constexpr int PIECES      = 4;     // 4 x b128 (16B) async loads per thread/tile
constexpr int TOPK        = 10;
constexpr int KLAB        = 5;

__global__ __launch_bounds__(TPB)
void embert_row_kernel(const float* __restrict__ probas,
                       const int*   __restrict__ labels,
                       float*       __restrict__ ws,
                       int V)
{
    __shared__ float buf[2][TILE_FLOATS];   // 32 KB double buffer
    __shared__ float mergebuf[TPB * TOPK];  // 10 KB merge scratch
    __shared__ float dummy16[4];            // sink for OOB async pieces
    __shared__ int   ulab_s[KLAB];
    __shared__ int   nc_s;
    __shared__ float csum_s;

    const int row = blockIdx.x;
    const int tid = threadIdx.x;
    const float* rowp = probas + (size_t)row * (size_t)V;

    // --- dedupe labels + correct-sum (thread 0; only <=5 scalar loads) -----
    if (tid == 0) {
        int nc = 0;
        for (int k = 0; k < KLAB; ++k) {
            int l = labels[row * KLAB + k];
            bool dup = false;
            for (int j = 0; j < nc; ++j) dup |= (ulab_s[j] == l);
            if (!dup) ulab_s[nc++] = l;
        }
        nc_s = nc;
        float cs = 0.f;
        for (int j = 0; j < nc; ++j) cs += rowp[ulab_s[j]];
        csum_s = cs;
    }
    __syncthreads();

    const int nc = nc_s;
    const int l0 = (0 < nc) ? ulab_s[0] : -1;
    const int l1 = (1 < nc) ? ulab_s[1] : -1;
    const int l2 = (2 < nc) ? ulab_s[2] : -1;
    const int l3 = (3 < nc) ? ulab_s[3] : -1;
    const int l4 = (4 < nc) ? ulab_s[4] : -1;

    // --- private sorted-descending top-10 in registers ---------------------
    float t10[TOPK];
#pragma unroll
    for (int k = 0; k < TOPK; ++k) t10[k] = -INFINITY;

    // Rare path: value beats current 10th best -> check labels, then insert.
    auto consider = [&](float v, int idx) {
        if (v > t10[TOPK - 1]) {
            bool is_lab = (idx == l0) | (idx == l1) | (idx == l2) |
                          (idx == l3) | (idx == l4);
            if (!is_lab) {
                float x = v;
#pragma unroll
                for (int k = 0; k < TOPK; ++k) {   // branchless sorted insert
                    float hi = fmaxf(t10[k], x);
                    x        = fminf(t10[k], x);
                    t10[k]   = hi;
                }
            }
        }
    };

    // Hot path: one max4 + one compare per 4 elements.
    auto consider4 = [&](float4 q, int fb) {
        float m = fmaxf(fmaxf(q.x, q.y), fmaxf(q.z, q.w));
        if (m > t10[TOPK - 1]) {
            consider(q.x, fb + 0);
            consider(q.y, fb + 1);
            consider(q.z, fb + 2);
            consider(q.w, fb + 3);
        }
    };

    // --- async tile issue: always exactly PIECES loads per thread ----------
    auto issue_tile = [&](int t, int b) {
#pragma unroll
        for (int j = 0; j < PIECES; ++j) {
            int pi = j * TPB + tid;                 // lane-coalesced pieces
            int fb = t * TILE_FLOATS + pi * 4;      // global float index
            bool ok = (fb + 4 <= V);
            unsigned lds = ok ? (unsigned)(uintptr_t)&buf[b][pi * 4]
                              : (unsigned)(uintptr_t)&dummy16[0];
            unsigned long long ga =
                (unsigned long long)(uintptr_t)(ok ? (rowp + fb) : rowp);
            // CDNA5 async DMA: memory -> LDS, ASYNCcnt-tracked, non-temporal
            // (single-pass stream larger than L2 -> don't retain).
            asm volatile("global_load_async_to_lds_b128 %0, %1, off th:TH_LOAD_NT"
                         :: "v"(lds), "v"(ga) : "memory");
        }
    };

    const int ntiles = (V + TILE_FLOATS - 1) / TILE_FLOATS;

    issue_tile(0, 0);
    for (int t = 0; t < ntiles; ++t) {
        const int cur = t & 1;
        if (t + 1 < ntiles) {
            issue_tile(t + 1, cur ^ 1);
            // 4 loads of tile t+1 may stay outstanding; tile t is complete
            asm volatile("s_wait_asynccnt 0x4" ::: "memory");
        } else {
            asm volatile("s_wait_asynccnt 0x0" ::: "memory");
        }

        // scan the pieces this thread loaded (no cross-thread LDS use)
#pragma unroll
        for (int j = 0; j < PIECES; ++j) {
            int pi = j * TPB + tid;
            int fb = t * TILE_FLOATS + pi * 4;
            if (fb + 4 <= V) {
                const float4 q =
                    *reinterpret_cast<const float4*>(&buf[cur][pi * 4]);
                consider4(q, fb);
            } else if (fb < V) {
                // straddling tail piece (<=1 per row): read global directly
                for (int e = 0; e < 4; ++e) {
                    int idx = fb + e;
                    if (idx < V) consider(rowp[idx], idx);
                }
            }
        }
    }

    // --- merge 256 sorted top-10 lists (log2 tree in LDS) ------------------
#pragma unroll
    for (int k = 0; k < TOPK; ++k) mergebuf[tid * TOPK + k] = t10[k];
    __syncthreads();

    for (int s = TPB / 2; s >= 1; s >>= 1) {
        if (tid < s) {
            float o[TOPK];
            const int ia = tid * TOPK;
            const int ib = (tid + s) * TOPK;
            int ca = 0, cb = 0;
#pragma unroll
            for (int k = 0; k < TOPK; ++k) {   // two-pointer merge, keep 10
                float av = mergebuf[ia + ca];
                float bv = mergebuf[ib + cb];
                bool  ta = (av >= bv);
                o[k] = ta ? av : bv;
                ca += ta ? 1 : 0;
                cb += ta ? 0 : 1;
            }
#pragma unroll
            for (int k = 0; k < TOPK; ++k) mergebuf[ia + k] = o[k];
        }
        __syncthreads();
    }

    if (tid == 0) {
        float wsum = 0.f;
#pragma unroll
        for (int k = 0; k < TOPK; ++k) wsum += mergebuf[k];
        const float wrong_avg   = wsum / (float)TOPK;
        const float correct_avg = csum_s / (float)nc;
        ws[row] = wrong_avg - correct_avg;
    }
}

// Deterministic final reduction: mean over B per-row partials -> scalar.
__global__ __launch_bounds__(TPB)
void embert_final_kernel(const float* __restrict__ ws,
                         float*       __restrict__ out,
                         int B)
{
    __shared__ float red[TPB];
    const int tid = threadIdx.x;
    float s = 0.f;
    for (int i = tid; i < B; i += TPB) s += ws[i];
    red[tid] = s;
    __syncthreads();
    for (int st = TPB / 2; st >= 1; st >>= 1) {
        if (tid < st) red[tid] += red[tid + st];
        __syncthreads();
    }
    if (tid == 0) out[0] = red[0] / (float)B;
}

extern "C" void kernel_launch(void* const* d_in, const int* in_sizes, int n_in,
                              void* d_out, int out_size, void* d_ws, size_t ws_size,
                              hipStream_t stream)
{
    const float* probas = (const float*)d_in[0];
    const int*   labels = (const int*)d_in[1];
    float*       out    = (float*)d_out;
    float*       ws     = (float*)d_ws;

    const int B = in_sizes[1] / KLAB;       // 2048
    const int V = in_sizes[0] / B;          // 50257

    embert_row_kernel<<<B, TPB, 0, stream>>>(probas, labels, ws, V);
    embert_final_kernel<<<1, TPB, 0, stream>>>(ws, out, B);
}